// GenerateProposals_10015863734532
// MI455X (gfx1250) — compile-verified
//
#include <hip/hip_runtime.h>

#define N_IMG   4
#define A_      15
#define H_      320
#define W_      320
#define HW_     (H_ * W_)          // 102400
#define AHW_    (A_ * HW_)         // 1536000
#define TILE_   4096
#define TPB_    256
#define TILES_  (AHW_ / TILE_)     // 375 (HW_/TILE_ = 25 exactly: tiles never straddle an a-plane)
#define PRE_TOPN  1000
#define POST_TOPN 100
#define CAND_CAP  4096
#define NMS_TH    0.7f
#define XFORM_CLIP 4.135166556742356f

// ---------- helpers: monotonic float <-> uint key (descending-friendly) ----------
__device__ __forceinline__ unsigned fkey(float f) {
    unsigned u = __float_as_uint(f);
    return (u & 0x80000000u) ? ~u : (u | 0x80000000u);
}
__device__ __forceinline__ float keyf(unsigned k) {
    unsigned u = (k & 0x80000000u) ? (k & 0x7FFFFFFFu) : ~k;
    return __uint_as_float(u);
}

// ---------- K0: zero histograms / counters (must run every call) ----------
__global__ void k0_zero(unsigned* __restrict__ p, int n) {
    int i = blockIdx.x * blockDim.x + threadIdx.x;
    if (i < n) p[i] = 0u;
}

// ---------- K1: decode boxes, validity-mask scores, emit keys + level-1 histogram ----------
__global__ void k1_decode(const float* __restrict__ scores,
                          const float* __restrict__ deltas,
                          const float* __restrict__ im_info,
                          const float* __restrict__ cell_anchors,
                          unsigned* __restrict__ keys,
                          unsigned* __restrict__ hist1)
{
    __shared__ float    s_sc[TILE_];
    __shared__ unsigned s_hist[4096];
    __shared__ float    s_ca[A_ * 4];

    const int tid  = threadIdx.x;
    const int n    = blockIdx.y;
    const long tile0 = (long)blockIdx.x * TILE_;

    for (int i = tid; i < 4096; i += TPB_) s_hist[i] = 0u;
    if (tid < A_ * 4) s_ca[tid] = cell_anchors[tid];

    // --- async stage this tile's scores into LDS (CDNA5 async path) ---
    const float* sbase = scores + (size_t)n * AHW_;
#pragma unroll
    for (int k = 0; k < 4; ++k) {
        int e = (k * TPB_ + tid) * 4;                         // 4 floats = 16B per lane
        unsigned lds_off = (unsigned)(size_t)(&s_sc[e]);      // low 32 bits of LDS flat addr = LDS offset
        unsigned goff    = (unsigned)((tile0 + e) * 4);
        asm volatile("global_load_async_to_lds_b128 %0, %1, %2"
                     :: "v"(lds_off), "v"(goff), "s"(sbase) : "memory");
    }

    const int a   = (int)(tile0 / HW_);
    const int hw0 = (int)(tile0 % HW_);
    const size_t dbase = (size_t)n * (A_ * 4) * HW_;
    __builtin_prefetch(deltas + dbase + (size_t)(a * 4) * HW_ + hw0, 0, 0);

    asm volatile("s_wait_asynccnt 0x0" ::: "memory");
    __syncthreads();

    const float imh = im_info[n * 3 + 0];
    const float imw = im_info[n * 3 + 1];
    const float ca0 = s_ca[a * 4 + 0], ca1 = s_ca[a * 4 + 1];
    const float ca2 = s_ca[a * 4 + 2], ca3 = s_ca[a * 4 + 3];
    const float aw  = ca2 - ca0, ahh = ca3 - ca1;
    const float cxo = 0.5f * (ca0 + ca2), cyo = 0.5f * (ca1 + ca3);

#pragma unroll
    for (int k = 0; k < 4; ++k) {
        int e  = (k * TPB_ + tid) * 4;
        int hw = hw0 + e;
        const float4 q0 = *reinterpret_cast<const float4*>(deltas + dbase + (size_t)(a * 4 + 0) * HW_ + hw);
        const float4 q1 = *reinterpret_cast<const float4*>(deltas + dbase + (size_t)(a * 4 + 1) * HW_ + hw);
        const float4 q2 = *reinterpret_cast<const float4*>(deltas + dbase + (size_t)(a * 4 + 2) * HW_ + hw);
        const float4 q3 = *reinterpret_cast<const float4*>(deltas + dbase + (size_t)(a * 4 + 3) * HW_ + hw);
        float dxv[4] = {q0.x, q0.y, q0.z, q0.w};
        float dyv[4] = {q1.x, q1.y, q1.z, q1.w};
        float dwv[4] = {q2.x, q2.y, q2.z, q2.w};
        float dhv[4] = {q3.x, q3.y, q3.z, q3.w};
#pragma unroll
        for (int m = 0; m < 4; ++m) {
            int wi = (hw + m) % W_;
            int hi = (hw + m) / W_;
            float acx = wi * 4.0f + cxo;
            float acy = hi * 4.0f + cyo;
            float dw  = fminf(dwv[m], XFORM_CLIP);
            float dh  = fminf(dhv[m], XFORM_CLIP);
            float pcx = dxv[m] * aw  + acx;
            float pcy = dyv[m] * ahh + acy;
            float pw  = __expf(dw) * aw;
            float ph  = __expf(dh) * ahh;
            float x1 = fminf(fmaxf(pcx - 0.5f * pw, 0.f), imw);
            float y1 = fminf(fmaxf(pcy - 0.5f * ph, 0.f), imh);
            float x2 = fminf(fmaxf(pcx + 0.5f * pw, 0.f), imw);
            float y2 = fminf(fmaxf(pcy + 0.5f * ph, 0.f), imh);
            bool valid = (x2 > x1) && (y2 > y1);
            float sc = valid ? s_sc[e + m] : -__builtin_inff();
            unsigned key = fkey(sc);
            keys[(size_t)n * AHW_ + tile0 + e + m] = key;
            atomicAdd(&s_hist[key >> 20], 1u);
        }
    }
    __syncthreads();
    for (int i = tid; i < 4096; i += TPB_) {
        unsigned c = s_hist[i];
        if (c) atomicAdd(&hist1[n * 4096 + i], c);
    }
}

// ---------- K2: find level-1 crossing bin ----------
__global__ void k2_thresh1(const unsigned* __restrict__ hist1, int* __restrict__ state) {
    int n = blockIdx.x;
    if (threadIdx.x == 0) {
        const unsigned* h = hist1 + n * 4096;
        unsigned cum = 0, above = 0;
        int b1 = 0; bool found = false;
        for (int b = 4095; b >= 0; --b) {
            unsigned c = h[b];
            if (!found && cum + c >= (unsigned)PRE_TOPN) { b1 = b; above = cum; found = true; }
            cum += c;
        }
        if (!found) { b1 = 0; above = cum - h[0]; }
        state[n * 8 + 0] = b1;
        state[n * 8 + 1] = (int)above;
    }
}

// ---------- K3: level-2 histogram restricted to crossing bin ----------
__global__ void k3_hist2(const unsigned* __restrict__ keys, const int* __restrict__ state,
                         unsigned* __restrict__ hist2)
{
    __shared__ unsigned s_hist[4096];
    const int tid = threadIdx.x;
    const int n   = blockIdx.y;
    const long tile0 = (long)blockIdx.x * TILE_;
    for (int i = tid; i < 4096; i += TPB_) s_hist[i] = 0u;
    __syncthreads();
    const unsigned B1 = (unsigned)state[n * 8 + 0];
    const unsigned* kp = keys + (size_t)n * AHW_ + tile0;
#pragma unroll 4
    for (int k = 0; k < 16; ++k) {
        unsigned key = kp[k * TPB_ + tid];
        if ((key >> 20) == B1) atomicAdd(&s_hist[(key >> 8) & 0xFFFu], 1u);
    }
    __syncthreads();
    for (int i = tid; i < 4096; i += TPB_) {
        unsigned c = s_hist[i];
        if (c) atomicAdd(&hist2[n * 4096 + i], c);
    }
}

// ---------- K4: finalize 24-bit threshold prefix ----------
__global__ void k4_thresh2(const unsigned* __restrict__ hist2, int* __restrict__ state) {
    int n = blockIdx.x;
    if (threadIdx.x == 0) {
        int b1    = state[n * 8 + 0];
        int above = state[n * 8 + 1];
        int need  = PRE_TOPN - above;          // > 0 by construction of K2
        const unsigned* h = hist2 + n * 4096;
        int cum = 0, b2 = 0; bool found = false;
        for (int b = 4095; b >= 0 && !found; --b) {
            cum += (int)h[b];
            if (cum >= need) { b2 = b; found = true; }
        }
        state[n * 8 + 2] = (b1 << 12) | b2;    // 24-bit prefix threshold
    }
}

// ---------- K5: compact candidates >= threshold prefix ----------
__global__ void k5_collect(const unsigned* __restrict__ keys, const int* __restrict__ state,
                           int* __restrict__ cnt, unsigned long long* __restrict__ cand)
{
    const int tid = threadIdx.x;
    const int n   = blockIdx.y;
    const long tile0 = (long)blockIdx.x * TILE_;
    const unsigned P = (unsigned)state[n * 8 + 2];
    const int a   = (int)(tile0 / HW_);
    const int hw0 = (int)(tile0 % HW_);
    const unsigned* kp = keys + (size_t)n * AHW_ + tile0;
#pragma unroll 4
    for (int k = 0; k < 16; ++k) {
        int e = k * TPB_ + tid;
        unsigned key = kp[e];
        if ((key >> 8) >= P) {
            int pos = atomicAdd(&cnt[n], 1);
            if (pos < CAND_CAP) {
                unsigned idx = (unsigned)((hw0 + e) * A_ + a);  // original flat anchor index
                cand[(size_t)n * CAND_CAP + pos] =
                    ((unsigned long long)key << 32) | (unsigned)(~idx); // desc key, asc idx ties
            }
        }
    }
}

// ---------- K6a: bitonic sort candidates; decode top-K boxes; pack finite flags ----------
__global__ __launch_bounds__(1024)
void k6a_sort_decode(const unsigned long long* __restrict__ cand, const int* __restrict__ cnt,
                     const float* __restrict__ deltas, const float* __restrict__ im_info,
                     const float* __restrict__ cell_anchors,
                     float4* __restrict__ boxes, float* __restrict__ probs,
                     unsigned* __restrict__ finw)
{
    __shared__ unsigned long long s_sort[CAND_CAP];
    const int tid = threadIdx.x;
    const int n   = blockIdx.x;

    int m = cnt[n]; if (m > CAND_CAP) m = CAND_CAP;
    for (int i = tid; i < CAND_CAP; i += 1024)
        s_sort[i] = (i < m) ? cand[(size_t)n * CAND_CAP + i] : 0ULL;
    __syncthreads();

    // bitonic sort, descending, 4096 elements
    for (int k = 2; k <= CAND_CAP; k <<= 1) {
        for (int j = k >> 1; j > 0; j >>= 1) {
            for (int t = tid; t < CAND_CAP / 2; t += 1024) {
                int ii = 2 * t - (t & (j - 1));
                int jj = ii + j;
                unsigned long long x = s_sort[ii], y = s_sort[jj];
                bool desc = ((ii & k) == 0);
                if ((x < y) == desc) { s_sort[ii] = y; s_sort[jj] = x; }
            }
            __syncthreads();
        }
    }

    const int K = (m < PRE_TOPN) ? m : PRE_TOPN;
    const size_t dbase = (size_t)n * (A_ * 4) * HW_;
    const float imh = im_info[n * 3 + 0];
    const float imw = im_info[n * 3 + 1];

    float4 ob = make_float4(0.f, 0.f, 0.f, 0.f);
    float  sc = 0.f;
    bool   finp = false;
    if (tid < K) {
        unsigned long long v = s_sort[tid];
        unsigned key = (unsigned)(v >> 32);
        unsigned idx = ~((unsigned)v);
        sc = keyf(key);
        int a = idx % A_, hw = idx / A_, wi = hw % W_, hi = hw / W_;
        float ca0 = cell_anchors[a * 4 + 0], ca1 = cell_anchors[a * 4 + 1];
        float ca2 = cell_anchors[a * 4 + 2], ca3 = cell_anchors[a * 4 + 3];
        float aw = ca2 - ca0, ahh = ca3 - ca1;
        float acx = wi * 4.0f + 0.5f * (ca0 + ca2);
        float acy = hi * 4.0f + 0.5f * (ca1 + ca3);
        float dx = deltas[dbase + (size_t)(a * 4 + 0) * HW_ + hw];
        float dy = deltas[dbase + (size_t)(a * 4 + 1) * HW_ + hw];
        float dw = fminf(deltas[dbase + (size_t)(a * 4 + 2) * HW_ + hw], XFORM_CLIP);
        float dh = fminf(deltas[dbase + (size_t)(a * 4 + 3) * HW_ + hw], XFORM_CLIP);
        float pcx = dx * aw + acx, pcy = dy * ahh + acy;
        float pw = __expf(dw) * aw, ph = __expf(dh) * ahh;
        ob.x = fminf(fmaxf(pcx - 0.5f * pw, 0.f), imw);
        ob.y = fminf(fmaxf(pcy - 0.5f * ph, 0.f), imh);
        ob.z = fminf(fmaxf(pcx + 0.5f * pw, 0.f), imw);
        ob.w = fminf(fmaxf(pcy + 0.5f * ph, 0.f), imh);
        finp = (key > 0x007FFFFFu);            // key(-inf) == 0x007FFFFF
    }
    boxes[n * 1024 + tid] = ob;
    probs[n * 1024 + tid] = finp ? sc : 0.f;
    // wave32: wave w covers candidates 32w..32w+31 -> one finite-word per wave
    unsigned long long bal = __ballot(finp ? 1 : 0);
    if ((tid & 31) == 0) finw[n * 32 + (tid >> 5)] = (unsigned)bal;
}

// ---------- K6b: parallel IoU suppression bitmask (j > i only) ----------
__global__ void k6b_mask(const float4* __restrict__ boxes, const int* __restrict__ cnt,
                         unsigned* __restrict__ mask)
{
    __shared__ float4 s_bx[1024];
    const int tid = threadIdx.x;
    const int n   = blockIdx.y;
    int K = cnt[n]; if (K > PRE_TOPN) K = PRE_TOPN;

    for (int i = tid; i < 1024; i += 256) s_bx[i] = boxes[n * 1024 + i];
    __syncthreads();

    const int r = blockIdx.x * 32 + (tid >> 3);
    if (r >= K) return;
    const float4 A = s_bx[r];
    const float areaA = (A.z - A.x) * (A.w - A.y);

    for (int w = (tid & 7); w < 32; w += 8) {
        unsigned bits = 0u;
        const int jbase = w << 5;
#pragma unroll 4
        for (int b = 0; b < 32; ++b) {
            int j = jbase + b;
            if (j > r && j < K) {
                float4 B = s_bx[j];
                float ix1 = fmaxf(A.x, B.x), iy1 = fmaxf(A.y, B.y);
                float ix2 = fminf(A.z, B.z), iy2 = fminf(A.w, B.w);
                float inter = fmaxf(ix2 - ix1, 0.f) * fmaxf(iy2 - iy1, 0.f);
                float uni = areaA + (B.z - B.x) * (B.w - B.y) - inter;
                float iou = (uni > 0.f) ? inter / fmaxf(uni, 1e-12f) : 0.f;
                if (iou > NMS_TH) bits |= (1u << b);
            }
        }
        mask[((size_t)n << 15) + (size_t)r * 32 + w] = bits;
    }
}

// ---------- K6c: single-wave greedy scan + rank + output (no barriers) ----------
__global__ void k6c_scan(const unsigned* __restrict__ mask, const unsigned* __restrict__ finw,
                         const float4* __restrict__ boxes, const float* __restrict__ probs,
                         const int* __restrict__ cnt,
                         float* __restrict__ out_rois, float* __restrict__ out_probs)
{
    const int n    = blockIdx.x;
    const int lane = threadIdx.x;           // 0..31, wave32
    int K = cnt[n]; if (K > PRE_TOPN) K = PRE_TOPN;

    unsigned supw = 0u;                     // lane L owns suppression bits for boxes 32L..32L+31
    for (int r0 = 0; r0 < K; r0 += 32) {
        unsigned tile[32];
#pragma unroll
        for (int t = 0; t < 32; ++t)        // batch-load 32 mask rows (pipelined, coalesced)
            tile[t] = (r0 + t < K) ? mask[((size_t)n << 15) + (size_t)(r0 + t) * 32 + lane] : 0u;
#pragma unroll
        for (int t = 0; t < 32; ++t) {
            int i = r0 + t;
            if (i < K) {
                unsigned wi = __shfl(supw, i >> 5, 32);      // ds_bpermute broadcast
                if (((wi >> (i & 31)) & 1u) == 0u) supw |= tile[t];
            }
        }
    }

    unsigned keepw = (~supw) & finw[n * 32 + lane];          // fin==0 beyond K
    int myc = __popc(keepw);
    int pre = 0, tot = 0;
    for (int k = 0; k < 32; ++k) {
        int v = __shfl(myc, k, 32);
        if (k < lane) pre += v;
        tot += v;
    }

    // zero the unfilled tail rows [min(tot,100), 100) -- disjoint from filled rows
    int filled = (tot < POST_TOPN) ? tot : POST_TOPN;
    for (int r = filled + lane; r < POST_TOPN; r += 32) {
        int row = n * POST_TOPN + r;
        out_rois[row * 5 + 0] = 0.f; out_rois[row * 5 + 1] = 0.f; out_rois[row * 5 + 2] = 0.f;
        out_rois[row * 5 + 3] = 0.f; out_rois[row * 5 + 4] = 0.f;
        out_probs[row] = 0.f;
    }

    unsigned kw = keepw;
    int rank = pre;
    while (kw) {
        int b = __ffs(kw) - 1;
        if (rank < POST_TOPN) {
            int j = (lane << 5) + b;
            float4 bx = boxes[n * 1024 + j];
            int row = n * POST_TOPN + rank;
            out_rois[row * 5 + 0] = (float)n;
            out_rois[row * 5 + 1] = bx.x;
            out_rois[row * 5 + 2] = bx.y;
            out_rois[row * 5 + 3] = bx.z;
            out_rois[row * 5 + 4] = bx.w;
            out_probs[row] = probs[n * 1024 + j];
        }
        ++rank;
        kw &= kw - 1;
    }
}

extern "C" void kernel_launch(void* const* d_in, const int* in_sizes, int n_in,
                              void* d_out, int out_size, void* d_ws, size_t ws_size,
                              hipStream_t stream)
{
    (void)in_sizes; (void)n_in; (void)out_size; (void)ws_size;
    const float* scores  = (const float*)d_in[0];
    const float* deltas  = (const float*)d_in[1];
    const float* im_info = (const float*)d_in[2];
    const float* anchors = (const float*)d_in[3];

    char* ws = (char*)d_ws;
    unsigned* keys  = (unsigned*)ws;                                   // N*AHW u32 = 24,576,000 B
    unsigned* hist1 = (unsigned*)(ws + (size_t)N_IMG * AHW_ * 4);      // N*4096 u32
    unsigned* hist2 = hist1 + N_IMG * 4096;                            // N*4096 u32
    int*      state = (int*)(hist2 + N_IMG * 4096);                    // N*8 i32
    int*      cnt   = state + N_IMG * 8;                               // N i32
    unsigned long long* cand = (unsigned long long*)(cnt + N_IMG);     // N*4096 u64 (16B-aligned)
    float4*   boxes_ws = (float4*)(cand + (size_t)N_IMG * CAND_CAP);   // N*1024 float4 (16B-aligned)
    float*    probs_ws = (float*)(boxes_ws + N_IMG * 1024);            // N*1024 f32
    unsigned* finw_ws  = (unsigned*)(probs_ws + N_IMG * 1024);         // N*32 u32
    unsigned* mask_ws  = finw_ws + N_IMG * 32;                         // N*1024*32 u32 = 512 KB

    // zero hist1..cnt (contiguous) every call (graph-replay safe)
    int zn = N_IMG * 4096 * 2 + N_IMG * 8 + N_IMG;
    k0_zero<<<(zn + 255) / 256, 256, 0, stream>>>(hist1, zn);

    dim3 grid(TILES_, N_IMG);
    k1_decode <<<grid, TPB_, 0, stream>>>(scores, deltas, im_info, anchors, keys, hist1);
    k2_thresh1<<<N_IMG, 32,  0, stream>>>(hist1, state);
    k3_hist2  <<<grid, TPB_, 0, stream>>>(keys, state, hist2);
    k4_thresh2<<<N_IMG, 32,  0, stream>>>(hist2, state);
    k5_collect<<<grid, TPB_, 0, stream>>>(keys, state, cnt, cand);

    float* out_rois  = (float*)d_out;
    float* out_probs = out_rois + N_IMG * POST_TOPN * 5;

    k6a_sort_decode<<<N_IMG, 1024, 0, stream>>>(cand, cnt, deltas, im_info, anchors,
                                                boxes_ws, probs_ws, finw_ws);
    k6b_mask<<<dim3(32, N_IMG), 256, 0, stream>>>(boxes_ws, cnt, mask_ws);
    k6c_scan<<<N_IMG, 32, 0, stream>>>(mask_ws, finw_ws, boxes_ws, probs_ws, cnt,
                                       out_rois, out_probs);
}